// Resample2d_40097814676027
// MI455X (gfx1250) — compile-verified
//
#include <hip/hip_runtime.h>
#include <hip/hip_bf16.h>

// Backward bilinear warp (resample2d), border clamp, align_corners semantics.
// input1: (B=8, C=64, H=256, W=448) fp32, NCHW
// input2: (B=8, 2,    H,     W)     fp32 flow (dx = plane 0, dy = plane 1)
// out:    (B, C, H, W) fp32
//
// One thread per output PIXEL (b,y,x); loops over all C channels so the flow
// read and bilinear-weight computation are amortized 64x. Memory-bound:
// ~478 MB compulsory traffic -> ~20 us roofline at 23.3 TB/s.
//
// Grid is 3D (x-chunk, y, b) so no div/mod is needed to recover coordinates.

namespace {
constexpr int B = 8;
constexpr int C = 64;
constexpr int H = 256;
constexpr int W = 448;
constexpr int HW = H * W;           // 114688
constexpr int XBLK = 224;           // 448 = 2 * 224; 224 = 7 wave32s
constexpr int CUNROLL = 8;
}

__global__ __launch_bounds__(XBLK, 8)
void Resample2d_40097814676027_kernel(const float* __restrict__ img,
                                      const float* __restrict__ flow,
                                      float* __restrict__ out) {
    // Coordinates straight from launch geometry: no integer division.
    const int x = blockIdx.x * XBLK + threadIdx.x;   // 0..447
    const int y = blockIdx.y;                        // 0..255
    const int b = blockIdx.z;                        // 0..7

    // Flow read: once per pixel, amortized over 64 channels.
    const int fbase = b * 2 * HW + y * W + x;
    const float fx = flow[fbase];
    const float fy = flow[fbase + HW];

    // Sample position, border-clamped (matches reference clip semantics).
    const float sx = fminf(fmaxf((float)x + fx, 0.0f), (float)(W - 1));
    const float sy = fminf(fmaxf((float)y + fy, 0.0f), (float)(H - 1));

    const float x0f = floorf(sx);
    const float y0f = floorf(sy);
    const float wx = sx - x0f;
    const float wy = sy - y0f;

    const int x0 = (int)x0f;
    const int y0 = (int)y0f;
    const int x1 = min(x0 + 1, W - 1);
    const int y1 = min(y0 + 1, H - 1);

    const float w00 = (1.0f - wx) * (1.0f - wy);
    const float w01 = wx * (1.0f - wy);
    const float w10 = (1.0f - wx) * wy;
    const float w11 = wx * wy;

    // Per-plane gather offsets (shared by all 64 channels of this batch).
    const int i00 = y0 * W + x0;
    const int i01 = y0 * W + x1;
    const int i10 = y1 * W + x0;
    const int i11 = y1 * W + x1;

    const float* __restrict__ ibase = img + (size_t)b * C * HW;
    float* __restrict__ obase       = out + (size_t)b * C * HW + (size_t)(y * W + x);

    // Channel loop: 4 cached gathers + 1 streaming (non-temporal) store each.
    // Unroll x8 -> 32 independent loads in flight; branchless clamped
    // prefetch of the next channel chunk (stride HW is fully predictable).
    #pragma unroll CUNROLL
    for (int c = 0; c < C; ++c) {
        const float* __restrict__ p = ibase + (size_t)c * HW;
        // Always-valid prefetch address (clamped to last channel): no branch.
        const int cpf = min(c + CUNROLL, C - 1);
        __builtin_prefetch(ibase + (size_t)cpf * HW + i00, 0, 1);

        const float v00 = p[i00];
        const float v01 = p[i01];
        const float v10 = p[i10];
        const float v11 = p[i11];
        const float v = v00 * w00 + v01 * w01 + v10 * w10 + v11 * w11;
        // Output is write-once streaming data: keep it out of L2 so input1
        // (235 MB, nearly fits the 192 MB L2) stays resident for the gathers.
        __builtin_nontemporal_store(v, obase + (size_t)c * HW);
    }
}

extern "C" void kernel_launch(void* const* d_in, const int* in_sizes, int n_in,
                              void* d_out, int out_size, void* d_ws, size_t ws_size,
                              hipStream_t stream) {
    (void)in_sizes; (void)n_in; (void)out_size; (void)d_ws; (void)ws_size;
    const float* img  = (const float*)d_in[0];
    const float* flow = (const float*)d_in[1];
    float* out        = (float*)d_out;

    dim3 block(XBLK, 1, 1);           // 7 wave32s
    dim3 grid(W / XBLK, H, B);        // (2, 256, 8)
    Resample2d_40097814676027_kernel<<<grid, block, 0, stream>>>(img, flow, out);
}